// TAM_26199300505931
// MI455X (gfx1250) — compile-verified
//
#include <hip/hip_runtime.h>
#include <math.h>

#define C_   12
#define T_   10
#define HW_  28
#define K_   3
#define TP   (T_ + 2)          // time-padded
#define TWOT 20                // 2*T
#define KPAD 12                // WMMA K padded to multiple of 4
#define NPAD 32                // WMMA N padded to two 16-wide tiles
#define EPS  1e-5f

typedef __attribute__((ext_vector_type(2))) float v2f;
typedef __attribute__((ext_vector_type(4))) float v4f;
typedef __attribute__((ext_vector_type(8))) float v8f;

__global__ __launch_bounds__(256)
void tam_fused_kernel(const float* __restrict__ x,
                      const float* __restrict__ w_g1,
                      const float* __restrict__ g1_gamma,
                      const float* __restrict__ g1_beta,
                      const float* __restrict__ g1_mean,
                      const float* __restrict__ g1_var,
                      const float* __restrict__ w_g2,
                      const float* __restrict__ w_l1,
                      const float* __restrict__ l1_gamma,
                      const float* __restrict__ l1_beta,
                      const float* __restrict__ l1_mean,
                      const float* __restrict__ l1_var,
                      const float* __restrict__ w_l2,
                      float* __restrict__ out)
{
    __shared__ float xs[C_ * T_ * HW_];      // 3360 floats: the whole sample
    __shared__ float pooledp[C_ * TP];       // time-padded pooled means (L branch)
    __shared__ float ash[16 * KPAD];         // zero-padded WMMA A tile (16 x 12)
    __shared__ float bsh[KPAD * NPAD];       // zero-padded WMMA B tile (12 x 32) = w_g1^T
    __shared__ float zs[C_ * TWOT];          // G-branch hidden (post BN+ReLU)
    __shared__ float as_[(C_ / 4) * T_];     // L-branch hidden (post BN+ReLU)
    __shared__ float actp[C_ * TP];          // time-padded sigmoid gate
    __shared__ float kern[C_ * K_];          // softmax temporal kernel

    const int tid = threadIdx.x;
    const int n   = blockIdx.x;

    // ---- Phase 0: stream sample into LDS (128-bit); stage padded WMMA tiles ----
    const v4f* xin = (const v4f*)(x + (size_t)n * (C_ * T_ * HW_));
    v4f* xs4 = (v4f*)xs;
    for (int i = tid; i < (C_ * T_ * HW_) / 4; i += 256) xs4[i] = xin[i];
    if (tid < C_) {
        pooledp[tid * TP]          = 0.0f;
        pooledp[tid * TP + T_ + 1] = 0.0f;
        actp[tid * TP]             = 0.0f;
        actp[tid * TP + T_ + 1]    = 0.0f;
    }
    // A tile: zero everything (valid entries overwritten in phase 1)
    if (tid < 16 * KPAD) ash[tid] = 0.0f;
    // B tile: bsh[k][j] = w_g1[j][k] for k<10, j<20 else 0  (j stride NPAD)
    {
        const int bi = tid - (16 * KPAD);           // threads 192..575 (here 192..255)
        // use a strided loop so 256 threads cover all 384 entries
        for (int i = tid; i < KPAD * NPAD; i += 256) {
            const int k = i / NPAD, j = i % NPAD;
            bsh[i] = (k < T_ && j < TWOT) ? w_g1[j * T_ + k] : 0.0f;
        }
        (void)bi;
    }
    __syncthreads();

    // ---- Phase 1: spatial mean pooling -> pooledp[c][1..T] and ash[c][t] ----
    if (tid < C_ * T_) {
        const int c = tid / T_, t = tid % T_;
        const float* row = xs + (c * T_ + t) * HW_;
        float s = 0.0f;
        #pragma unroll
        for (int k = 0; k < HW_; ++k) s += row[k];
        s *= (1.0f / HW_);
        pooledp[c * TP + t + 1] = s;
        ash[c * KPAD + t]       = s;       // A(m=c, k=t)
    }
    __syncthreads();

    const int wave = tid >> 5;
    if (wave == 0) {
        // ---- Phase 2a (wave 0): z = pooled(12x10) @ w_g1^T(10x20), WMMA f32 16x16x4
        // A 16x4: lane -> M (l&15); VGPR r + lane-half -> K = kq + r
        // B 4x16: lane -> N (l&15); VGPR r + lane-half -> K = kq + r
        const int l  = tid & 31;
        const int m  = l & 15;
        const int kq = (l >> 4) * 2;
        v8f d0 = {0.f,0.f,0.f,0.f,0.f,0.f,0.f,0.f};
        v8f d1 = {0.f,0.f,0.f,0.f,0.f,0.f,0.f,0.f};
        #pragma unroll
        for (int kb = 0; kb < KPAD; kb += 4) {
            const int k0i = kb + kq;               // k for r=0; r=1 is k0i+1 (< KPAD)
            v2f a, b0, b1;
            a[0]  = ash[m * KPAD + k0i];
            a[1]  = ash[m * KPAD + k0i + 1];
            b0[0] = bsh[k0i * NPAD + m];
            b0[1] = bsh[(k0i + 1) * NPAD + m];
            b1[0] = bsh[k0i * NPAD + m + 16];
            b1[1] = bsh[(k0i + 1) * NPAD + m + 16];
            d0 = __builtin_amdgcn_wmma_f32_16x16x4_f32(false, a, false, b0,
                                                       (short)0, d0, false, false);
            d1 = __builtin_amdgcn_wmma_f32_16x16x4_f32(false, a, false, b1,
                                                       (short)0, d1, false, false);
        }
        // BN + ReLU straight out of the accumulators, scatter to zs
        #pragma unroll
        for (int r = 0; r < 8; ++r) {
            const int mm = (l < 16) ? r : r + 8;   // row (channel)
            const int j0 = l & 15;                 // column
            if (mm < C_) {
                float v = (d0[r] - g1_mean[j0]) * rsqrtf(g1_var[j0] + EPS)
                          * g1_gamma[j0] + g1_beta[j0];
                zs[mm * TWOT + j0] = fmaxf(v, 0.0f);
                if (j0 < 4) {
                    const int j = j0 + 16;
                    float w = (d1[r] - g1_mean[j]) * rsqrtf(g1_var[j] + EPS)
                              * g1_gamma[j] + g1_beta[j];
                    zs[mm * TWOT + j] = fmaxf(w, 0.0f);
                }
            }
        }
    } else if (wave == 1) {
        // ---- Phase 2b (wave 1): L-branch conv1 (C->C/4, K=3, pad 1) + BN + ReLU
        const int idx = tid - 32;
        if (idx < (C_ / 4) * T_) {
            const int c4 = idx / T_, t = idx % T_;
            float s = 0.0f;
            for (int ci = 0; ci < C_; ++ci) {
                #pragma unroll
                for (int k = 0; k < K_; ++k)
                    s += w_l1[(c4 * C_ + ci) * K_ + k] * pooledp[ci * TP + t + k];
            }
            s = (s - l1_mean[c4]) * rsqrtf(l1_var[c4] + EPS) * l1_gamma[c4] + l1_beta[c4];
            as_[idx] = fmaxf(s, 0.0f);
        }
    }
    __syncthreads();

    // ---- Phase 3a: logits (12x20 @ 20x3) + softmax -> kern ----
    if (tid < C_) {
        const int c = tid;
        float lg[K_];
        #pragma unroll
        for (int j = 0; j < K_; ++j) {
            float s = 0.0f;
            for (int q = 0; q < TWOT; ++q) s += zs[c * TWOT + q] * w_g2[j * TWOT + q];
            lg[j] = s;
        }
        const float mx = fmaxf(fmaxf(lg[0], lg[1]), lg[2]);
        float e0 = expf(lg[0] - mx), e1 = expf(lg[1] - mx), e2 = expf(lg[2] - mx);
        const float inv = 1.0f / (e0 + e1 + e2);
        kern[c * K_ + 0] = e0 * inv;
        kern[c * K_ + 1] = e1 * inv;
        kern[c * K_ + 2] = e2 * inv;
    }
    // ---- Phase 3b: pointwise conv (C/4->C) + sigmoid -> actp ----
    if (tid >= 128 && tid < 128 + C_ * T_) {
        const int idx = tid - 128;
        const int c = idx / T_, t = idx % T_;
        float s = 0.0f;
        #pragma unroll
        for (int c4 = 0; c4 < C_ / 4; ++c4)
            s += w_l2[c * (C_ / 4) + c4] * as_[c4 * T_ + t];
        actp[c * TP + t + 1] = 1.0f / (1.0f + expf(-s));
    }
    __syncthreads();

    // ---- Phase 4: modulate + depthwise temporal conv + (t,c) transpose store ----
    // out[t][c][s] = sum_j kern[c][j] * act[c][t+j-1] * x[c][t+j-1][s]
    // actp pad zeros make boundary coefficients zero; LDS time index is clamped.
    v4f* out4 = (v4f*)(out + (size_t)n * (T_ * C_ * HW_));
    for (int i = tid; i < T_ * C_ * (HW_ / 4); i += 256) {
        const int q = i % (HW_ / 4);
        const int c = (i / (HW_ / 4)) % C_;
        const int t = i / (HW_ / 4) / C_;
        const float k0 = kern[c * K_ + 0] * actp[c * TP + t];       // act[c][t-1]
        const float k1 = kern[c * K_ + 1] * actp[c * TP + t + 1];   // act[c][t]
        const float k2 = kern[c * K_ + 2] * actp[c * TP + t + 2];   // act[c][t+1]
        const int tm = (t > 0) ? t - 1 : 0;
        const int tq = (t < T_ - 1) ? t + 1 : T_ - 1;
        const v4f a0 = xs4[(c * T_ + tm) * (HW_ / 4) + q];
        const v4f a1 = xs4[(c * T_ + t ) * (HW_ / 4) + q];
        const v4f a2 = xs4[(c * T_ + tq) * (HW_ / 4) + q];
        v4f r;
        r.x = k0 * a0.x + k1 * a1.x + k2 * a2.x;
        r.y = k0 * a0.y + k1 * a1.y + k2 * a2.y;
        r.z = k0 * a0.z + k1 * a1.z + k2 * a2.z;
        r.w = k0 * a0.w + k1 * a1.w + k2 * a2.w;
        out4[(t * C_ + c) * (HW_ / 4) + q] = r;
    }
}

extern "C" void kernel_launch(void* const* d_in, const int* in_sizes, int n_in,
                              void* d_out, int out_size, void* d_ws, size_t ws_size,
                              hipStream_t stream) {
    const float* x        = (const float*)d_in[0];
    const float* w_g1     = (const float*)d_in[1];
    const float* g1_gamma = (const float*)d_in[2];
    const float* g1_beta  = (const float*)d_in[3];
    const float* g1_mean  = (const float*)d_in[4];
    const float* g1_var   = (const float*)d_in[5];
    const float* w_g2     = (const float*)d_in[6];
    const float* w_l1     = (const float*)d_in[7];
    const float* l1_gamma = (const float*)d_in[8];
    const float* l1_beta  = (const float*)d_in[9];
    const float* l1_mean  = (const float*)d_in[10];
    const float* l1_var   = (const float*)d_in[11];
    const float* w_l2     = (const float*)d_in[12];
    float* out            = (float*)d_out;

    const int ntot = in_sizes[0] / (C_ * T_ * HW_);
    tam_fused_kernel<<<ntot, 256, 0, stream>>>(x, w_g1, g1_gamma, g1_beta, g1_mean,
                                               g1_var, w_g2, w_l1, l1_gamma, l1_beta,
                                               l1_mean, l1_var, w_l2, out);
}